// corr_meta_learning_62500364091655
// MI455X (gfx1250) — compile-verified
//
#include <hip/hip_runtime.h>
#include <hip/hip_bf16.h>

typedef __attribute__((ext_vector_type(2))) float v2f;
typedef __attribute__((ext_vector_type(8))) float v8f;

#define B_N   16384
#define F_N   39
#define E_N   16
#define VOCAB 26000
#define P_N   741
#define INDIM 2223      // 39*38 + 741
#define HSTR  2224      // padded row stride for h (even -> aligned b64 loads)
#define WOFFN 1482
#define H1S   112       // padded 100 -> 112 (7 tiles of 16)
#define LR_C  0.001f

// ---------------------------------------------------------------------------
// Multi-value butterfly: reduce 8 per-lane values across each 16-lane half.
// On return, lane l holds the full 16-lane sum of value index (l & 7).
// 8 shuffles total (vs 32 for naive per-register trees).
// ---------------------------------------------------------------------------
__device__ __forceinline__ float rowdot16(float p0, float p1, float p2, float p3,
                                          float p4, float p5, float p6, float p7,
                                          int l)
{
    float a0 = (l & 1) ? p1 : p0, a1 = (l & 1) ? p0 : p1;
    a0 += __shfl_xor(a1, 1);
    float b0 = (l & 1) ? p3 : p2, b1 = (l & 1) ? p2 : p3;
    b0 += __shfl_xor(b1, 1);
    float c0 = (l & 1) ? p5 : p4, c1 = (l & 1) ? p4 : p5;
    c0 += __shfl_xor(c1, 1);
    float d0 = (l & 1) ? p7 : p6, d1 = (l & 1) ? p6 : p7;
    d0 += __shfl_xor(d1, 1);
    float e0 = (l & 2) ? b0 : a0, e1 = (l & 2) ? a0 : b0;
    e0 += __shfl_xor(e1, 2);
    float f0 = (l & 2) ? d0 : c0, f1 = (l & 2) ? c0 : d0;
    f0 += __shfl_xor(f1, 2);
    float g0 = (l & 4) ? f0 : e0, g1 = (l & 4) ? e0 : f0;
    g0 += __shfl_xor(g1, 4);
    g0 += __shfl_xor(g0, 8);
    return g0;
}

// ---------------------------------------------------------------------------
// K1: gather embeddings, per-sample meta-learning step, sparse projection,
//     write woff part of h (cols 0..1481) + zero pad col, vx, lin sums.
// ---------------------------------------------------------------------------
__global__ void k_meta(const int* __restrict__ x, const float* __restrict__ emb,
                       const float* __restrict__ linw, const float* __restrict__ w0,
                       float* __restrict__ vx, float* __restrict__ h,
                       float* __restrict__ lin)
{
    __shared__ float vxs[F_N * E_N];     // 624
    __shared__ float w0s[F_N * F_N];     // 1521
    __shared__ float w1s[F_N * F_N];     // 1521
    __shared__ float wvs[F_N * E_N];     // 624 (wv, then wv2)
    __shared__ float dmus[F_N];
    __shared__ float mus[F_N];
    __shared__ int   idxs[F_N];
    __shared__ float linsum;

    const int b = blockIdx.x, t = threadIdx.x;

    if (t < F_N) idxs[t] = x[b * F_N + t] + t * VOCAB;
    if (t == 0) linsum = 0.0f;
    __syncthreads();

    for (int i = t; i < F_N * E_N; i += 128) {
        const int f = i >> 4, e = i & 15;
        const float v = emb[(long)idxs[f] * E_N + e];
        vxs[i] = v;
        vx[(long)b * (F_N * E_N) + i] = v;
    }
    for (int i = t; i < F_N * F_N; i += 128) w0s[i] = w0[i];
    if (t < F_N) atomicAdd(&linsum, linw[idxs[t]]);
    __syncthreads();
    if (t == 0) { lin[b] = linsum; h[(long)b * HSTR + INDIM] = 0.0f; }

    // wv = w0 @ vx
    for (int i = t; i < F_N * E_N; i += 128) {
        const int r = i >> 4, e = i & 15;
        float s = 0.0f;
        for (int j = 0; j < F_N; ++j) s += w0s[r * F_N + j] * vxs[j * E_N + e];
        wvs[i] = s;
    }
    __syncthreads();

    // w1 = w0 - (LR/F)*(wv @ vx^T); diag -> -1
    const float c = LR_C / (float)F_N;
    for (int i = t; i < F_N * F_N; i += 128) {
        const int r = i / F_N, j = i % F_N;
        float s = 0.0f;
        #pragma unroll
        for (int e = 0; e < E_N; ++e) s += wvs[r * E_N + e] * vxs[j * E_N + e];
        w1s[i] = (r == j) ? -1.0f : (w0s[i] - c * s);
    }
    __syncthreads();

    // wv2 = w1 @ vx (reuse wvs)
    for (int i = t; i < F_N * E_N; i += 128) {
        const int r = i >> 4, e = i & 15;
        float s = 0.0f;
        for (int j = 0; j < F_N; ++j) s += w1s[r * F_N + j] * vxs[j * E_N + e];
        wvs[i] = s;
    }
    __syncthreads();

    if (t < F_N) {
        float s = 0.0f;
        #pragma unroll
        for (int e = 0; e < E_N; ++e) { const float v = wvs[t * E_N + e]; s += v * v; }
        dmus[t] = 0.5f * s;
    }
    __syncthreads();

    if (t == 0) {   // simplex projection, 39 elems
        float z[F_N], zs[F_N];
        for (int i = 0; i < F_N; ++i) { z[i] = (1.0f / F_N) - LR_C * dmus[i]; zs[i] = z[i]; }
        for (int i = 1; i < F_N; ++i) {
            const float key = zs[i]; int j = i - 1;
            while (j >= 0 && zs[j] < key) { zs[j + 1] = zs[j]; --j; }
            zs[j + 1] = key;
        }
        const float z0 = zs[0];
        float cs = 0.0f, tau = 0.0f;
        for (int i = 0; i < F_N; ++i) {
            const float v = zs[i] - z0; cs += v;
            const float ta = (cs - 1.0f) / (float)(i + 1);
            if (v > ta) tau = ta;
        }
        for (int i = 0; i < F_N; ++i) {
            const float v = (z[i] - z0) - tau;
            mus[i] = v > 0.0f ? v : 0.0f;
        }
    }
    __syncthreads();

    for (int m = t; m < WOFFN; m += 128) {
        const int flat = 1 + (m / 39) * 40 + (m % 39);
        const int r = flat / F_N;
        h[(long)b * HSTR + m] = mus[r] * w1s[flat];
    }
}

// ---------------------------------------------------------------------------
// K2: pair interaction via f32 WMMA. One block per pair, K[p] in LDS.
// Each wave runs 2 sample-tiles per loop (2 accumulators share B operands),
// then a single 8-shuffle butterfly per tile for the row-wise dot.
// ---------------------------------------------------------------------------
__global__ void k_pairs(const float* __restrict__ vx, const float* __restrict__ kern,
                        float* __restrict__ h)
{
    __shared__ float Ks[E_N * E_N];
    __shared__ int rc[2];
    const int p = blockIdx.x, t = threadIdx.x;

    if (t == 0) {
        int i = 0, rem = p;
        while (rem >= 38 - i) { rem -= 38 - i; ++i; }
        rc[0] = i; rc[1] = i + 1 + rem;
    }
    Ks[t] = kern[p * 256 + t];
    __syncthreads();

    const int rowF = rc[0], colF = rc[1];
    const int lane = t & 31, wid = t >> 5;
    const int m = lane & 15, hi = lane >> 4;

    for (int tile = wid * 2; tile < B_N / 16; tile += 16) {
        const int sb0 = tile * 16, sb1 = sb0 + 16;
        const float* ar0 = vx + (long)(sb0 + m) * (F_N * E_N) + rowF * E_N + 2 * hi;
        const float* ar1 = vx + (long)(sb1 + m) * (F_N * E_N) + rowF * E_N + 2 * hi;
        v8f acc0 = {}, acc1 = {};
        #pragma unroll
        for (int kc = 0; kc < 4; ++kc) {
            v2f bm; bm.x = Ks[(4 * kc + hi) * 16 + m];
                    bm.y = Ks[(4 * kc + 2 + hi) * 16 + m];
            v2f a0 = *(const v2f*)(ar0 + 4 * kc);
            v2f a1 = *(const v2f*)(ar1 + 4 * kc);
            acc0 = __builtin_amdgcn_wmma_f32_16x16x4_f32(false, a0, false, bm,
                                                         (short)0, acc0, false, false);
            acc1 = __builtin_amdgcn_wmma_f32_16x16x4_f32(false, a1, false, bm,
                                                         (short)0, acc1, false, false);
        }
        float p0[8], p1[8];
        #pragma unroll
        for (int r = 0; r < 8; ++r) {
            const int mm = r + 8 * hi;
            p0[r] = acc0[r] * vx[(long)(sb0 + mm) * (F_N * E_N) + colF * E_N + m];
            p1[r] = acc1[r] * vx[(long)(sb1 + mm) * (F_N * E_N) + colF * E_N + m];
        }
        const float s0 = rowdot16(p0[0], p0[1], p0[2], p0[3], p0[4], p0[5], p0[6], p0[7], lane);
        const float s1 = rowdot16(p1[0], p1[1], p1[2], p1[3], p1[4], p1[5], p1[6], p1[7], lane);
        if ((lane & 8) == 0) {
            const int mm = (lane & 7) + 8 * hi;
            h[(long)(sb0 + mm) * HSTR + WOFFN + p] = s0;
            h[(long)(sb1 + mm) * HSTR + WOFFN + p] = s1;
        }
    }
}

// ---------------------------------------------------------------------------
// K3: BN1 column stats over B -> affine (A, Bc) per column.
// ---------------------------------------------------------------------------
__global__ void k_bn1(const float* __restrict__ h, const float* __restrict__ g,
                      const float* __restrict__ be, float* __restrict__ bnA,
                      float* __restrict__ bnB)
{
    const int col = blockIdx.x * 256 + threadIdx.x;
    if (col >= INDIM) return;
    float s = 0.0f, ss = 0.0f;
    for (int b = 0; b < B_N; ++b) {
        const float v = h[(long)b * HSTR + col];
        s += v; ss += v * v;
    }
    const float mean = s * (1.0f / B_N);
    const float var  = ss * (1.0f / B_N) - mean * mean;
    const float A = g[col] * rsqrtf(var + 1e-5f);
    bnA[col] = A;
    bnB[col] = be[col] - mean * A;
}

// ---------------------------------------------------------------------------
// Kp1: fold BN1 into fc1 -> W1p (112x2223, rows>=100 zero), b1p (112).
// ---------------------------------------------------------------------------
__global__ void k_fold1(const float* __restrict__ fc1w, const float* __restrict__ fc1b,
                        const float* __restrict__ bnA, const float* __restrict__ bnB,
                        float* __restrict__ W1p, float* __restrict__ b1p)
{
    __shared__ float red[256];
    const int o = blockIdx.x, t = threadIdx.x;
    float s = 0.0f;
    if (o < 100) {
        for (int cc = t; cc < INDIM; cc += 256) {
            const float w = fc1w[o * INDIM + cc];
            W1p[(long)o * INDIM + cc] = bnA[cc] * w;
            s += bnB[cc] * w;
        }
    } else {
        for (int cc = t; cc < INDIM; cc += 256) W1p[(long)o * INDIM + cc] = 0.0f;
    }
    red[t] = s; __syncthreads();
    for (int off = 128; off > 0; off >>= 1) {
        if (t < off) red[t] += red[t + off];
        __syncthreads();
    }
    if (t == 0) b1p[o] = (o < 100) ? fc1b[o] + red[0] : 0.0f;
}

// ---------------------------------------------------------------------------
// K4: fc1 GEMM (16384x2224)@(2224x112) fused bias+ReLU, f32 WMMA.
// 8 waves/block share one 2224x16 B strip in LDS; each wave holds TWO
// accumulators (2 M-tiles) so every B chunk feeds 2 WMMAs.
// ---------------------------------------------------------------------------
__global__ void k_gemm1(const float* __restrict__ h, const float* __restrict__ W1p,
                        const float* __restrict__ b1p, float* __restrict__ h1)
{
    extern __shared__ float Bls[];               // [2224][16]
    const int t = threadIdx.x;
    const int nt = blockIdx.x % 7;
    const int mg = blockIdx.x / 7;               // 0..63

    {   // stage W1p^T strip: Bls[k*16+n] = W1p[(nt*16+n)*2223 + k], pad k=2223
        const int n = t >> 4, k0 = t & 15;
        const float* src = W1p + (long)(nt * 16 + n) * INDIM;
        for (int k = k0; k < HSTR; k += 16)
            Bls[k * 16 + n] = (k < INDIM) ? src[k] : 0.0f;
    }
    __syncthreads();

    const int lane = t & 31, wid = t >> 5;
    const int m = lane & 15, hi = lane >> 4;
    const int mt = (mg * 8 + wid) * 2;           // 0..1022 (2 tiles per wave)
    const long rb0 = (long)(mt * 16 + m) * HSTR;
    const long rb1 = rb0 + 16L * HSTR;

    v8f acc0 = {}, acc1 = {};
    for (int k = 0; k < HSTR; k += 4) {
        if ((k & 63) == 0) {
            __builtin_prefetch(&h[rb0 + k + 128], 0, 1);
            __builtin_prefetch(&h[rb1 + k + 128], 0, 1);
        }
        const int kg = k + 2 * hi;
        const v2f a0 = *(const v2f*)(h + rb0 + kg);
        const v2f a1 = *(const v2f*)(h + rb1 + kg);
        v2f bm;
        bm.x = Bls[(k + hi)     * 16 + m];
        bm.y = Bls[(k + 2 + hi) * 16 + m];
        acc0 = __builtin_amdgcn_wmma_f32_16x16x4_f32(false, a0, false, bm,
                                                     (short)0, acc0, false, false);
        acc1 = __builtin_amdgcn_wmma_f32_16x16x4_f32(false, a1, false, bm,
                                                     (short)0, acc1, false, false);
    }

    const float bb = b1p[nt * 16 + m];
    #pragma unroll
    for (int r = 0; r < 8; ++r) {
        const int mm = r + 8 * hi;
        const float v0 = acc0[r] + bb;
        const float v1 = acc1[r] + bb;
        h1[(long)(mt * 16 + mm) * H1S + nt * 16 + m]        = v0 > 0.0f ? v0 : 0.0f;
        h1[(long)((mt + 1) * 16 + mm) * H1S + nt * 16 + m]  = v1 > 0.0f ? v1 : 0.0f;
    }
}

// ---------------------------------------------------------------------------
// K5: BN2 stats on h1 cols 0..99.
// ---------------------------------------------------------------------------
__global__ void k_bn2(const float* __restrict__ h1, const float* __restrict__ g,
                      const float* __restrict__ be, float* __restrict__ bnA,
                      float* __restrict__ bnB)
{
    const int col = threadIdx.x;
    if (col >= H1S) return;
    if (col >= 100) { bnA[col] = 0.0f; bnB[col] = 0.0f; return; }
    float s = 0.0f, ss = 0.0f;
    for (int b = 0; b < B_N; ++b) {
        const float v = h1[(long)b * H1S + col];
        s += v; ss += v * v;
    }
    const float mean = s * (1.0f / B_N);
    const float var  = ss * (1.0f / B_N) - mean * mean;
    const float A = g[col] * rsqrtf(var + 1e-5f);
    bnA[col] = A;
    bnB[col] = be[col] - mean * A;
}

// ---------------------------------------------------------------------------
// Kp2: fold BN2 into fc2 -> W2p (112x112 zero-padded), b2p (112).
// ---------------------------------------------------------------------------
__global__ void k_fold2(const float* __restrict__ fc2w, const float* __restrict__ fc2b,
                        const float* __restrict__ bnA, const float* __restrict__ bnB,
                        float* __restrict__ W2p, float* __restrict__ b2p)
{
    __shared__ float red[128];
    const int o = blockIdx.x, t = threadIdx.x;
    float s = 0.0f;
    if (t < H1S) {
        const int cc = t;
        const float w = (o < 100 && cc < 100) ? fc2w[o * 100 + cc] : 0.0f;
        W2p[o * H1S + cc] = (cc < 100) ? bnA[cc] * w : 0.0f;
        if (cc < 100) s += bnB[cc] * w;
    }
    red[t] = s; __syncthreads();
    for (int off = 64; off > 0; off >>= 1) {
        if (t < off) red[t] += red[t + off];
        __syncthreads();
    }
    if (t == 0) b2p[o] = (o < 100) ? fc2b[o] + red[0] : 0.0f;
}

// ---------------------------------------------------------------------------
// K6: fc2 GEMM (16384x112)@(112x112) fused bias+ReLU, f32 WMMA, 2 M-tiles/wave.
// ---------------------------------------------------------------------------
__global__ void k_gemm2(const float* __restrict__ h1, const float* __restrict__ W2p,
                        const float* __restrict__ b2p, float* __restrict__ h2)
{
    __shared__ float Bls[H1S * 16];              // 7 KB
    const int t = threadIdx.x;
    const int nt = blockIdx.x % 7;
    const int mg = blockIdx.x / 7;               // 0..63

    {   // Bls[k*16+n] = W2p[(nt*16+n)*112 + k]
        const int n = t >> 4, k0 = t & 15;
        const float* src = W2p + (nt * 16 + n) * H1S;
        for (int k = k0; k < H1S; k += 16) Bls[k * 16 + n] = src[k];
    }
    __syncthreads();

    const int lane = t & 31, wid = t >> 5;
    const int m = lane & 15, hi = lane >> 4;
    const int mt = (mg * 8 + wid) * 2;
    const long rb0 = (long)(mt * 16 + m) * H1S;
    const long rb1 = rb0 + 16L * H1S;

    v8f acc0 = {}, acc1 = {};
    #pragma unroll 4
    for (int k = 0; k < H1S; k += 4) {
        const int kg = k + 2 * hi;
        const v2f a0 = *(const v2f*)(h1 + rb0 + kg);
        const v2f a1 = *(const v2f*)(h1 + rb1 + kg);
        v2f bm;
        bm.x = Bls[(k + hi)     * 16 + m];
        bm.y = Bls[(k + 2 + hi) * 16 + m];
        acc0 = __builtin_amdgcn_wmma_f32_16x16x4_f32(false, a0, false, bm,
                                                     (short)0, acc0, false, false);
        acc1 = __builtin_amdgcn_wmma_f32_16x16x4_f32(false, a1, false, bm,
                                                     (short)0, acc1, false, false);
    }

    const float bb = b2p[nt * 16 + m];
    #pragma unroll
    for (int r = 0; r < 8; ++r) {
        const int mm = r + 8 * hi;
        const float v0 = acc0[r] + bb;
        const float v1 = acc1[r] + bb;
        h2[(long)(mt * 16 + mm) * H1S + nt * 16 + m]       = v0 > 0.0f ? v0 : 0.0f;
        h2[(long)((mt + 1) * 16 + mm) * H1S + nt * 16 + m] = v1 > 0.0f ? v1 : 0.0f;
    }
}

// ---------------------------------------------------------------------------
// K7: fc3 (100 live cols) + bias + lin.
// ---------------------------------------------------------------------------
__global__ void k_final(const float* __restrict__ h2, const float* __restrict__ fc3w,
                        const float* __restrict__ fc3b, const float* __restrict__ lin,
                        float* __restrict__ out)
{
    __shared__ float w3[100];
    const int t = threadIdx.x;
    if (t < 100) w3[t] = fc3w[t];
    __syncthreads();
    const int b = blockIdx.x * 256 + t;
    const float* row = h2 + (long)b * H1S;
    float s = fc3b[0] + lin[b];
    for (int cc = 0; cc < 100; ++cc) s += row[cc] * w3[cc];
    out[b] = s;
}

// ---------------------------------------------------------------------------
extern "C" void kernel_launch(void* const* d_in, const int* in_sizes, int n_in,
                              void* d_out, int out_size, void* d_ws, size_t ws_size,
                              hipStream_t stream)
{
    const int*   x     = (const int*)  d_in[0];
    const float* emb   = (const float*)d_in[1];
    const float* linw  = (const float*)d_in[2];
    const float* w0    = (const float*)d_in[3];
    const float* kern  = (const float*)d_in[4];
    const float* bn1g  = (const float*)d_in[5];
    const float* bn1b  = (const float*)d_in[6];
    const float* fc1w  = (const float*)d_in[7];
    const float* fc1b  = (const float*)d_in[8];
    const float* bn2g  = (const float*)d_in[9];
    const float* bn2b  = (const float*)d_in[10];
    const float* fc2w  = (const float*)d_in[11];
    const float* fc2b  = (const float*)d_in[12];
    const float* fc3w  = (const float*)d_in[13];
    const float* fc3b  = (const float*)d_in[14];
    float* out = (float*)d_out;

    float* ws   = (float*)d_ws;
    float* vx   = ws;                              // B*624
    float* h    = vx   + (long)B_N * F_N * E_N;    // B*2224 (padded)
    float* lin  = h    + (long)B_N * HSTR;         // B
    float* h1   = lin  + B_N;                      // B*112
    float* h2   = h1   + (long)B_N * H1S;          // B*112
    float* bn1A = h2   + (long)B_N * H1S;          // 2223
    float* bn1B = bn1A + INDIM;
    float* bn2A = bn1B + INDIM;                    // 112
    float* bn2B = bn2A + H1S;
    float* W1p  = bn2B + H1S;                      // 112*2223
    float* b1p  = W1p  + (long)H1S * INDIM;        // 112
    float* W2p  = b1p  + H1S;                      // 112*112
    float* b2p  = W2p  + H1S * H1S;                // 112

    k_meta <<<B_N, 128, 0, stream>>>(x, emb, linw, w0, vx, h, lin);
    k_pairs<<<P_N, 256, 0, stream>>>(vx, kern, h);
    k_bn1  <<<(INDIM + 255) / 256, 256, 0, stream>>>(h, bn1g, bn1b, bn1A, bn1B);
    k_fold1<<<H1S, 256, 0, stream>>>(fc1w, fc1b, bn1A, bn1B, W1p, b1p);
    k_gemm1<<<7 * 64, 256, HSTR * 16 * sizeof(float), stream>>>(h, W1p, b1p, h1);
    k_bn2  <<<1, H1S, 0, stream>>>(h1, bn2g, bn2b, bn2A, bn2B);
    k_fold2<<<H1S, 128, 0, stream>>>(fc2w, fc2b, bn2A, bn2B, W2p, b2p);
    k_gemm2<<<7 * 64, 256, 0, stream>>>(h1, W2p, b2p, h2);
    k_final<<<B_N / 256, 256, 0, stream>>>(h2, fc3w, fc3b, lin, out);
}